// DinkNet_dgl_22608707846325
// MI455X (gfx1250) — compile-verified
//
#include <hip/hip_runtime.h>

typedef float v2f __attribute__((ext_vector_type(2)));
typedef float v8f __attribute__((ext_vector_type(8)));

constexpr int HD   = 96;   // hidden dim
constexpr int RPB  = 128;  // rows per block in GEMM (8 waves x 16 rows)
constexpr int BK   = 32;   // K-tile staged in LDS
constexpr int LPAD = 36;   // X-tile LDS row stride (144B: 16B aligned, conflict-free A reads)
constexpr int WPAD = 38;   // W-tile LDS row stride (152B: 8B aligned b64, conflict-free B reads)

// ---------------- utility kernels ----------------

__global__ void fill_zero_kernel(float* __restrict__ p, long n) {
  long i = (long)blockIdx.x * blockDim.x + threadIdx.x;
  long stride = (long)gridDim.x * blockDim.x;
  for (; i < n; i += stride) p[i] = 0.0f;
}

__global__ void degree_kernel(const int* __restrict__ src, const int* __restrict__ dst,
                              float* __restrict__ deg_out, float* __restrict__ deg_in, int E) {
  int e = blockIdx.x * blockDim.x + threadIdx.x;
  if (e < E) {
    __hip_atomic_fetch_add(&deg_out[src[e]], 1.0f, __ATOMIC_RELAXED, __HIP_MEMORY_SCOPE_AGENT);
    __hip_atomic_fetch_add(&deg_in[dst[e]],  1.0f, __ATOMIC_RELAXED, __HIP_MEMORY_SCOPE_AGENT);
  }
}

__global__ void finalize_norm_kernel(float* __restrict__ p, int n) {
  int i = blockIdx.x * blockDim.x + threadIdx.x;
  if (i < n) p[i] = 1.0f / sqrtf(fmaxf(p[i], 1.0f));
}

// wsum[k] = sum_j Wm[k][j]; bsum = sum_j bm[j]  (collapses final GEMM+rowsum to a matvec)
__global__ void wsum_kernel(const float* __restrict__ Wm, const float* __restrict__ bm,
                            float* __restrict__ wsum, float* __restrict__ bsum) {
  int t = threadIdx.x;
  if (t < HD) {
    float s = 0.0f;
    for (int j = 0; j < HD; ++j) s += Wm[t * HD + j];
    wsum[t] = s;
  } else if (t == HD) {
    float s = 0.0f;
    for (int j = 0; j < HD; ++j) s += bm[j];
    *bsum = s;
  }
}

// ---------------- fused gather * norm GEMM via V_WMMA_F32_16X16X4_F32 ----------------
// OUT[r, :] = (X[rowmap ? rowmap[r] : r, :] * norm[r]) @ W      (OUT is [nrows, 96])
// K must be a multiple of 4 (500 and 96 both are).
// Both operands staged in LDS: steady-state loop is pipelined ds_load_b64 + v_wmma.
__global__ __launch_bounds__(256)
void gemm_norm_wmma_kernel(const float* __restrict__ X, const int* __restrict__ rowmap,
                           const float* __restrict__ norm, const float* __restrict__ W,
                           float* __restrict__ OUT, int nrows, int K) {
  __shared__ float ldsX[RPB][LPAD];   // [row_in_tile][kk]        (~18.0 KB)
  __shared__ float ldsW[HD][WPAD];    // [col][kk] (transposed)   (~14.3 KB)

  const int tid  = threadIdx.x;
  const int wave = tid >> 5;
  const int lane = tid & 31;
  const int half = lane >> 4;   // 0: lanes 0-15, 1: lanes 16-31
  const int l16  = lane & 15;
  const int row0 = blockIdx.x * RPB;

  v8f acc[6] = {};  // 6 col-tiles of 16 => 96 columns, one A-fragment reused 6x

  for (int k0 = 0; k0 < K; k0 += BK) {
    // ---- stage RPB x BK tile of (X * norm) into LDS, coalesced float4 loads ----
    #pragma unroll
    for (int it = 0; it < (RPB * BK / 4) / 256; ++it) {   // 4 iters
      int idx  = tid + 256 * it;       // 0..1023
      int r    = idx >> 3;             // 8 float4-chunks per row
      int c    = idx & 7;
      int grow = row0 + r;
      int kg   = k0 + c * 4;
      float4 v = make_float4(0.f, 0.f, 0.f, 0.f);
      if (grow < nrows && kg < K) {
        int srow = rowmap ? rowmap[grow] : grow;
        const float* p = X + (long)srow * K + kg;
        v = *(const float4*)p;
        if (kg + BK < K) __builtin_prefetch(p + BK, 0, 0);  // next K-tile -> global_prefetch
        float s = norm[grow];
        v.x *= s; v.y *= s; v.z *= s; v.w *= s;
      }
      ldsX[r][c * 4 + 0] = v.x;
      ldsX[r][c * 4 + 1] = v.y;
      ldsX[r][c * 4 + 2] = v.z;
      ldsX[r][c * 4 + 3] = v.w;
    }
    // ---- stage BK x 96 tile of W into LDS, transposed to [col][kk] ----
    #pragma unroll
    for (int it = 0; it < (BK * HD) / 256; ++it) {        // 12 iters
      int idx = tid + 256 * it;        // 0..3071
      int kk  = idx / HD;              // 0..31
      int col = idx % HD;              // 0..95 (consecutive tid -> consecutive col: coalesced)
      float v = 0.0f;
      if (k0 + kk < K) v = W[(long)(k0 + kk) * HD + col];
      ldsW[col][kk] = v;
    }
    __syncthreads();

    const int kmax = (BK < K - k0) ? BK : (K - k0);
    const int mrow = wave * 16 + l16;  // A-frag: M = lane % 16 (both halves)
    for (int kk = 0; kk < kmax; kk += 4) {
      int ka = kk + half * 2;          // even -> 8B aligned

      // Issue ALL fragment loads first (distinct registers) so ds_loads pipeline
      // behind the WMMA pipe instead of a full s_wait_dscnt 0 per matrix op.
      v2f A = *(const v2f*)&ldsX[mrow][ka];
      v2f Bf[6];
      #pragma unroll
      for (int t = 0; t < 6; ++t)
        Bf[t] = *(const v2f*)&ldsW[t * 16 + l16][ka];

      #pragma unroll
      for (int t = 0; t < 6; ++t)
        acc[t] = __builtin_amdgcn_wmma_f32_16x16x4_f32(
            /*neg_a=*/false, A, /*neg_b=*/false, Bf[t],
            /*c_mod=*/(short)0, acc[t], /*reuse_a=*/false, /*reuse_b=*/false);
    }
    __syncthreads();
  }

  // ---- store: C/D layout: VGPR i -> M = i (+8 for lanes 16-31), N = lane%16 ----
  const int orow = row0 + wave * 16 + half * 8;
  #pragma unroll
  for (int t = 0; t < 6; ++t) {
    int col = t * 16 + l16;
    #pragma unroll
    for (int i = 0; i < 8; ++i) {
      int r = orow + i;
      if (r < nrows) OUT[(long)r * HD + col] = acc[t][i];
    }
  }
}

// ---------------- SpMM: agg[dst] += h[src], one wave per edge ----------------
__global__ void spmm_kernel(const int* __restrict__ src, const int* __restrict__ dst,
                            const float* __restrict__ h, float* __restrict__ agg, int E) {
  int gw   = (int)(((long)blockIdx.x * blockDim.x + threadIdx.x) >> 5);
  int lane = threadIdx.x & 31;
  if (gw < E) {
    int s = src[gw], d = dst[gw];
    const float* hp = h + (long)s * HD;
    float* ap = agg + (long)d * HD;
    #pragma unroll
    for (int i = 0; i < 3; ++i) {
      float v = hp[lane + 32 * i];
      __hip_atomic_fetch_add(&ap[lane + 32 * i], v, __ATOMIC_RELAXED, __HIP_MEMORY_SCOPE_AGENT);
    }
  }
}

// ---------------- epilogue: out = PReLU(agg * norm_in + b) ----------------
__global__ void prelu_kernel(const float* __restrict__ agg, const float* __restrict__ norm_in,
                             const float* __restrict__ b, const float* __restrict__ a,
                             float* __restrict__ out, int n) {
  long idx = (long)blockIdx.x * blockDim.x + threadIdx.x;
  if (idx < (long)n * HD) {
    int i = (int)(idx / HD);
    int j = (int)(idx % HD);
    float v = agg[idx] * norm_in[i] + b[j];
    out[idx] = (v >= 0.0f) ? v : a[j] * v;
  }
}

// ---------------- final matvec reduction: out[r] = dot(z[r,:], wsum) + bsum ----------------
__global__ void reduce_kernel(const float* __restrict__ z, const float* __restrict__ wsum,
                              const float* __restrict__ bsum, float* __restrict__ out, int n) {
  int gw   = (int)(((long)blockIdx.x * blockDim.x + threadIdx.x) >> 5);
  int lane = threadIdx.x & 31;
  if (gw < n) {
    const float* zp = z + (long)gw * HD;
    float s = 0.0f;
    #pragma unroll
    for (int i = 0; i < 3; ++i) s = fmaf(zp[lane + 32 * i], wsum[lane + 32 * i], s);
    #pragma unroll
    for (int off = 16; off > 0; off >>= 1) s += __shfl_xor(s, off, 32);
    if (lane == 0) out[gw] = s + *bsum;
  }
}

// ---------------- host launch ----------------
extern "C" void kernel_launch(void* const* d_in, const int* in_sizes, int n_in,
                              void* d_out, int out_size, void* d_ws, size_t ws_size,
                              hipStream_t stream) {
  const float* x    = (const float*)d_in[0];
  const int*   src  = (const int*)  d_in[1];
  const int*   dst  = (const int*)  d_in[2];
  const int*   perm = (const int*)  d_in[3];
  const float* W1   = (const float*)d_in[4];
  const float* b1   = (const float*)d_in[5];
  const float* a1   = (const float*)d_in[6];
  const float* W2   = (const float*)d_in[7];
  const float* b2   = (const float*)d_in[8];
  const float* a2   = (const float*)d_in[9];
  const float* Wm   = (const float*)d_in[10];
  const float* bm   = (const float*)d_in[11];
  float* out = (float*)d_out;

  const int E  = in_sizes[1];
  const int N  = in_sizes[3];
  const int K1 = in_sizes[0] / N;   // 500

  float* ws       = (float*)d_ws;
  float* norm_out = ws;                      // [N]
  float* norm_in  = ws + N;                  // [N]  (contiguous with norm_out)
  float* wsum     = ws + 2L * N;             // [96]
  float* bsum     = ws + 2L * N + HD;        // [1]
  float* bufA     = ws + 2L * N + 2 * HD;    // [N*96], 16B-aligned offset
  float* bufB     = bufA + (long)N * HD;     // [N*96]

  const long NH = (long)N * HD;
  const int gemm_blocks = (N + RPB - 1) / RPB;
  const int spmm_blocks = (int)(((long)E * 32 + 255) / 256);
  const int elw_blocks  = (int)((NH + 255) / 256);
  const int red_blocks  = (int)(((long)N * 32 + 255) / 256);

  // degrees -> norms (shared by both views)
  fill_zero_kernel<<<512, 256, 0, stream>>>(norm_out, 2L * N);
  degree_kernel<<<(E + 255) / 256, 256, 0, stream>>>(src, dst, norm_out, norm_in, E);
  finalize_norm_kernel<<<(2 * N + 255) / 256, 256, 0, stream>>>(norm_out, 2 * N);
  wsum_kernel<<<1, 128, 0, stream>>>(Wm, bm, wsum, bsum);

  for (int view = 0; view < 2; ++view) {
    const int* rowmap = view ? perm : nullptr;
    // layer 1: (gather(x) * norm_out) @ W1 -> SpMM -> PReLU
    gemm_norm_wmma_kernel<<<gemm_blocks, 256, 0, stream>>>(x, rowmap, norm_out, W1, bufA, N, K1);
    fill_zero_kernel<<<2048, 256, 0, stream>>>(bufB, NH);
    spmm_kernel<<<spmm_blocks, 256, 0, stream>>>(src, dst, bufA, bufB, E);
    prelu_kernel<<<elw_blocks, 256, 0, stream>>>(bufB, norm_in, b1, a1, bufA, N);
    // layer 2
    gemm_norm_wmma_kernel<<<gemm_blocks, 256, 0, stream>>>(bufA, nullptr, norm_out, W2, bufB, N, HD);
    fill_zero_kernel<<<2048, 256, 0, stream>>>(bufA, NH);
    spmm_kernel<<<spmm_blocks, 256, 0, stream>>>(src, dst, bufB, bufA, E);
    prelu_kernel<<<elw_blocks, 256, 0, stream>>>(bufA, norm_in, b2, a2, bufB, N);
    // projection + row-sum collapsed into matvec
    reduce_kernel<<<red_blocks, 256, 0, stream>>>(bufB, wsum, bsum, out + (long)view * N, N);
  }
}